// EntropyKD_17480516894990
// MI455X (gfx1250) — compile-verified
//
#include <hip/hip_runtime.h>
#include <math.h>

#define B_    8
#define C_    1024
#define L_    4096
#define NPB   (C_ * L_)        // 4,194,304 elements per batch
#define NCOL  (B_ * L_)        // 32,768 columns
#define K1BPB 64               // blocks per batch for stats
#define K3BLK 256
#define K3GRD 128              // 128 blocks x 64 col-groups-of-4 = 32768 columns

typedef float v2f __attribute__((ext_vector_type(2)));
typedef float v4f __attribute__((ext_vector_type(4)));
typedef float v8f __attribute__((ext_vector_type(8)));

// Deterministic full-wave (32 lane) sum using V_WMMA_F32_16X16X4_F32.
// A: lane L<16 -> A[L,0]=v, A[L,1]=0 ; lane L>=16 -> A[L-16,2]=v, A[L-16,3]=0.
// B = ones(4x16)  =>  D[m,n] = v_m + v_{m+16} (independent of n).
// Lanes 0-15 hold rows 0-7 in their 8 D VGPRs, lanes 16-31 hold rows 8-15,
// so (sum of own 8 rows) + shfl_xor(16) = total over all 32 lanes.
__device__ __forceinline__ float wave_sum32(float v) {
  v2f a; a[0] = v;    a[1] = 0.0f;
  v2f b; b[0] = 1.0f; b[1] = 1.0f;
  v8f c = {};
  v8f d = __builtin_amdgcn_wmma_f32_16x16x4_f32(
      /*neg_a=*/false, a, /*neg_b=*/false, b,
      /*c_mod=*/(short)0, c, /*reuse_a=*/false, /*reuse_b=*/false);
  float s = d[0] + d[1] + d[2] + d[3] + d[4] + d[5] + d[6] + d[7];
  s += __shfl_xor(s, 16, 32);
  return s;
}

// ---- Kernel 1: per-batch partial sum / sum-of-squares of x_st (double) ----
__global__ void k_batch_stats(const float* __restrict__ xt, double2* __restrict__ bp) {
  const int b = blockIdx.y;
  const v4f* p = (const v4f*)(xt + (size_t)b * NPB);   // 16B-aligned: NPB*4 % 16 == 0
  double s = 0.0, q = 0.0;
  for (int i = blockIdx.x * blockDim.x + threadIdx.x; i < NPB / 4;
       i += gridDim.x * blockDim.x) {
    v4f v = p[i];
#pragma unroll
    for (int j = 0; j < 4; ++j) {
      double d = (double)v[j];
      s += d;
      q += d * d;
    }
  }
  __shared__ double ss[256];
  __shared__ double sq[256];
  ss[threadIdx.x] = s; sq[threadIdx.x] = q;
  __syncthreads();
  for (int st = 128; st > 0; st >>= 1) {
    if (threadIdx.x < st) {
      ss[threadIdx.x] += ss[threadIdx.x + st];
      sq[threadIdx.x] += sq[threadIdx.x + st];
    }
    __syncthreads();
  }
  if (threadIdx.x == 0)
    bp[b * K1BPB + blockIdx.x] = make_double2(ss[0], sq[0]);
}

// ---- Kernel 2: finalize margin[b] (ddof=1 std, norm.cdf formula) ----
__global__ void k_margin(const double2* __restrict__ bp, float* __restrict__ margin) {
  const int b = blockIdx.x;
  __shared__ double ss[64];
  __shared__ double sq[64];
  double2 v = bp[b * K1BPB + threadIdx.x];
  ss[threadIdx.x] = v.x; sq[threadIdx.x] = v.y;
  __syncthreads();
  for (int st = 32; st > 0; st >>= 1) {
    if (threadIdx.x < st) {
      ss[threadIdx.x] += ss[threadIdx.x + st];
      sq[threadIdx.x] += sq[threadIdx.x + st];
    }
    __syncthreads();
  }
  if (threadIdx.x == 0) {
    const double N    = (double)NPB;
    const double mean = ss[0] / N;
    const double var  = (sq[0] - ss[0] * ss[0] / N) / (N - 1.0);
    const double sd   = sqrt(fabs(var));
    const double r    = mean / sd;                     // z = -r
    const double cdf  = 0.5 * (1.0 + erf(-r / 1.4142135623730951));
    double m;
    if (cdf > 0.001) {
      const double scdf = (cdf < 1e-30) ? 1e-30 : cdf;
      m = -sd * exp(-0.5 * r * r) / 2.5066282746310002 / scdf + mean;
    } else {
      m = -3.0 * sd;
    }
    margin[b] = (float)m;
  }
}

// ---- Kernel 3: column softmax / log-softmax entropy ------------------------
// Each thread owns 4 adjacent l-columns (b128 loads) and a 256-channel slice
// of C (4-way C split). Per-slice online-softmax states are merged via LDS,
// then pass B writes entropy and accumulates the loss partial.
__global__ void __launch_bounds__(K3BLK)
k_entropy(const float* __restrict__ xs, const float* __restrict__ xt,
          const float* __restrict__ margin, float* __restrict__ out,
          float* __restrict__ lpart) {
  const int tid      = threadIdx.x;
  const int cchunk   = tid >> 6;                   // 0..3 -> channels [256*cchunk, +256)
  const int gidx     = tid & 63;                   // col-group within block
  const int colgroup = blockIdx.x * 64 + gidx;     // 0..8191 (groups of 4 columns)
  const int b        = colgroup >> 10;             // / (L_/4)
  const int lbase    = (colgroup & 1023) << 2;     // 4 adjacent l values
  const size_t base  = (size_t)b * NPB + lbase;
  const v4f* ps = (const v4f*)(xs + base);         // row stride L_/4 in v4f units
  const v4f* pt = (const v4f*)(xt + base);
  const float mg = margin[b];
  const int c0 = cchunk << 8;

  // Pass A: online max + sum-exp over this thread's 256 channels, 4 columns.
  float msS[4], zsS[4], msT[4], zsT[4];
#pragma unroll
  for (int j = 0; j < 4; ++j) { msS[j] = -INFINITY; zsS[j] = 0.0f;
                                msT[j] = -INFINITY; zsT[j] = 0.0f; }
  for (int c = 0; c < 256; ++c) {
    const v4f vs = ps[(size_t)(c0 + c) * (L_ / 4)];
    const v4f vt = pt[(size_t)(c0 + c) * (L_ / 4)];
#pragma unroll
    for (int j = 0; j < 4; ++j) {
      const float s = vs[j];
      if (s > msS[j]) { zsS[j] = zsS[j] * expf(msS[j] - s) + 1.0f; msS[j] = s; }
      else            { zsS[j] += expf(s - msS[j]); }
      const float t = fmaxf(vt[j], mg);
      if (t > msT[j]) { zsT[j] = zsT[j] * expf(msT[j] - t) + 1.0f; msT[j] = t; }
      else            { zsT[j] += expf(t - msT[j]); }
    }
  }

  // Merge the 4 C-slices of each column group through LDS (deterministic).
  __shared__ v4f shSm[256], shSz[256], shTm[256], shTz[256];
  v4f tmpa, tmpb, tmpc, tmpd;
#pragma unroll
  for (int j = 0; j < 4; ++j) { tmpa[j] = msS[j]; tmpb[j] = zsS[j];
                                tmpc[j] = msT[j]; tmpd[j] = zsT[j]; }
  const int slot = (gidx << 2) | cchunk;
  shSm[slot] = tmpa; shSz[slot] = tmpb; shTm[slot] = tmpc; shTz[slot] = tmpd;
  __syncthreads();

  float Ms[4], lzs[4], Mt[4], lzt[4];
#pragma unroll
  for (int j = 0; j < 4; ++j) { Ms[j] = -INFINITY; Mt[j] = -INFINITY; }
  for (int k = 0; k < 4; ++k) {
    const v4f a = shSm[(gidx << 2) | k];
    const v4f c2 = shTm[(gidx << 2) | k];
#pragma unroll
    for (int j = 0; j < 4; ++j) { Ms[j] = fmaxf(Ms[j], a[j]); Mt[j] = fmaxf(Mt[j], c2[j]); }
  }
  float Zs[4] = {0.f, 0.f, 0.f, 0.f}, Zt[4] = {0.f, 0.f, 0.f, 0.f};
  for (int k = 0; k < 4; ++k) {
    const v4f a = shSm[(gidx << 2) | k];
    const v4f z = shSz[(gidx << 2) | k];
    const v4f c2 = shTm[(gidx << 2) | k];
    const v4f y = shTz[(gidx << 2) | k];
#pragma unroll
    for (int j = 0; j < 4; ++j) {
      Zs[j] += z[j] * expf(a[j] - Ms[j]);
      Zt[j] += y[j] * expf(c2[j] - Mt[j]);
    }
  }
#pragma unroll
  for (int j = 0; j < 4; ++j) { lzs[j] = logf(Zs[j]); lzt[j] = logf(Zt[j]); }

  // Pass B: entropy = softmax(xs) * log_softmax(target); store + partial sum.
  float acc = 0.0f;
  float* po = out + 1 + base;                      // d_out[0] is the loss scalar
  for (int c = 0; c < 256; ++c) {
    const int cc = c0 + c;
    const v4f vs = ps[(size_t)cc * (L_ / 4)];
    const v4f vt = pt[(size_t)cc * (L_ / 4)];
#pragma unroll
    for (int j = 0; j < 4; ++j) {
      const float e = expf(vs[j] - Ms[j] - lzs[j]) *
                      ((fmaxf(vt[j], mg) - Mt[j]) - lzt[j]);
      po[(size_t)cc * L_ + j] = e;
      acc += e;
    }
  }

  // Deterministic block reduction: WMMA wave-sum, then 8 wave partials via LDS.
  const float w = wave_sum32(acc);
  __shared__ float sw[8];
  const int lane = threadIdx.x & 31;
  const int wid  = threadIdx.x >> 5;
  if (lane == 0) sw[wid] = w;
  __syncthreads();
  if (threadIdx.x == 0) {
    float t = 0.0f;
    for (int i = 0; i < 8; ++i) t += sw[i];
    lpart[blockIdx.x] = t;
  }
}

// ---- Kernel 4: final loss = -sum(entropy) / (B*L) ----
__global__ void k_loss(const float* __restrict__ lpart, float* __restrict__ out) {
  float v = (threadIdx.x < K3GRD) ? lpart[threadIdx.x] : 0.0f;
  const float w = wave_sum32(v);
  __shared__ float sw[8];
  const int lane = threadIdx.x & 31;
  const int wid  = threadIdx.x >> 5;
  if (lane == 0) sw[wid] = w;
  __syncthreads();
  if (threadIdx.x == 0) {
    float t = 0.0f;
    for (int i = 0; i < 8; ++i) t += sw[i];
    out[0] = -t / (float)NCOL;
  }
}

extern "C" void kernel_launch(void* const* d_in, const int* in_sizes, int n_in,
                              void* d_out, int out_size, void* d_ws, size_t ws_size,
                              hipStream_t stream) {
  (void)in_sizes; (void)n_in; (void)out_size; (void)ws_size;
  const float* x_ts = (const float*)d_in[2];   // source  [8,1024,64,64]
  const float* x_st = (const float*)d_in[3];   // target  [8,1024,64,64]
  float* out = (float*)d_out;                  // [0]=loss, [1..]=entropy

  char* ws = (char*)d_ws;
  double2* bp     = (double2*)ws;              // 8*64 double2 = 8 KB
  float*   margin = (float*)(ws + 8192);       // 8 floats
  float*   lpart  = (float*)(ws + 8192 + 256); // 128 floats

  k_batch_stats<<<dim3(K1BPB, B_), 256, 0, stream>>>(x_st, bp);
  k_margin<<<B_, 64, 0, stream>>>(bp, margin);
  k_entropy<<<K3GRD, K3BLK, 0, stream>>>(x_ts, x_st, margin, out, lpart);
  k_loss<<<1, 256, 0, stream>>>(lpart, out);
}